// LinearAttention_5858335392404
// MI455X (gfx1250) — compile-verified
//
#include <hip/hip_runtime.h>
#include <math.h>

#define BATCH 16
#define DIMC  256
#define RED   32
#define NPIX  16384   // 128*128

typedef float v2f __attribute__((ext_vector_type(2)));
typedef float v8f __attribute__((ext_vector_type(8)));

static __device__ __forceinline__ v8f wmma4(v2f a, v2f b, v8f c) {
    // V_WMMA_F32_16X16X4_F32 : D = A(16x4,f32) * B(4x16,f32) + C(16x16,f32)
    return __builtin_amdgcn_wmma_f32_16x16x4_f32(
        /*neg_a=*/false, a, /*neg_b=*/false, b,
        /*c_mod=*/(short)0, c, /*reuse_a=*/false, /*reuse_b=*/false);
}

// CDNA5 async copy: global memory -> LDS, 16 bytes per lane, tracked by ASYNCcnt.
static __device__ __forceinline__ void async_load_b128(unsigned lds_off, const float* gaddr) {
    asm volatile("global_load_async_to_lds_b128 %0, %1, off"
                 :: "v"(lds_off), "v"(gaddr) : "memory");
}
static __device__ __forceinline__ void wait_async0() {
#if __has_builtin(__builtin_amdgcn_s_wait_asynccnt)
    __builtin_amdgcn_s_wait_asynccnt(0);
#else
    asm volatile("s_wait_asynccnt 0x0" ::: "memory");
#endif
}

// ---------------------------------------------------------------- kernel 0
__global__ void zero_f32(float* __restrict__ p, int n) {
    int i = blockIdx.x * blockDim.x + threadIdx.x;
    if (i < n) p[i] = 0.0f;
}

// ---------------------------------------------------------------- kernel 1
// k[b,r,n] = sum_c k_w[r,c]*y[b,c,n] + k_b[r]
// grid (NPIX/512, BATCH), block 256 (8 waves). wave covers 32(M) x 64(N).
__global__ void k_project(const float* __restrict__ y,
                          const float* __restrict__ kw,
                          const float* __restrict__ kb,
                          float* __restrict__ kws) {
    const int lane = threadIdx.x & 31;
    const int wave = threadIdx.x >> 5;
    const int l16  = lane & 15;
    const int half = lane >> 4;
    const int b    = blockIdx.y;
    const int n0   = blockIdx.x * 512 + wave * 64;

    const float* yb = y + (size_t)b * DIMC * NPIX;

    v8f acc[2][4];
    #pragma unroll
    for (int mt = 0; mt < 2; ++mt)
        #pragma unroll
        for (int nt = 0; nt < 4; ++nt) acc[mt][nt] = {};

    for (int c0 = 0; c0 < DIMC; c0 += 4) {
        v2f af[2];
        #pragma unroll
        for (int mt = 0; mt < 2; ++mt) {
            const float* ap = kw + (size_t)(mt * 16 + l16) * DIMC + c0 + 2 * half;
            af[mt].x = ap[0]; af[mt].y = ap[1];
        }
        v2f bf[4];
        #pragma unroll
        for (int nt = 0; nt < 4; ++nt) {
            const float* bp = yb + (size_t)(c0 + 2 * half) * NPIX + n0 + nt * 16 + l16;
            bf[nt].x = bp[0]; bf[nt].y = bp[NPIX];
        }
        #pragma unroll
        for (int mt = 0; mt < 2; ++mt)
            #pragma unroll
            for (int nt = 0; nt < 4; ++nt)
                acc[mt][nt] = wmma4(af[mt], bf[nt], acc[mt][nt]);
    }

    #pragma unroll
    for (int mt = 0; mt < 2; ++mt)
        #pragma unroll
        for (int nt = 0; nt < 4; ++nt)
            #pragma unroll
            for (int vi = 0; vi < 8; ++vi) {
                int r   = mt * 16 + vi + 8 * half;
                int col = n0 + nt * 16 + l16;
                kws[((size_t)b * RED + r) * NPIX + col] = acc[mt][nt][vi] + kb[r];
            }
}

// ---------------------------------------------------------------- kernel 2
// per-(b,r) row: max and sum(exp(k-max)) over 16384 elements
__global__ void softmax_stats(const float* __restrict__ kws,
                              float* __restrict__ mx,
                              float* __restrict__ sm) {
    __shared__ float sred[256];
    const int row = blockIdx.x;           // b*RED + r
    const int tid = threadIdx.x;
    const float* p = kws + (size_t)row * NPIX;

    float m = -INFINITY;
    for (int i = tid; i < NPIX; i += 256) m = fmaxf(m, p[i]);
    sred[tid] = m; __syncthreads();
    for (int s = 128; s > 0; s >>= 1) {
        if (tid < s) sred[tid] = fmaxf(sred[tid], sred[tid + s]);
        __syncthreads();
    }
    const float rowmax = sred[0];
    __syncthreads();

    float acc = 0.0f;
    for (int i = tid; i < NPIX; i += 256) acc += __expf(p[i] - rowmax);
    sred[tid] = acc; __syncthreads();
    for (int s = 128; s > 0; s >>= 1) {
        if (tid < s) sred[tid] += sred[tid + s];
        __syncthreads();
    }
    if (tid == 0) { mx[row] = rowmax; sm[row] = sred[0]; }
}

// ---------------------------------------------------------------- kernel 2.5
// kws[b,r,n] <- exp(kws[b,r,n] - mx[b,r])   (in place, float4 vectorized)
__global__ void exp_inplace(float* __restrict__ kws, const float* __restrict__ mx) {
    size_t i4 = ((size_t)blockIdx.x * blockDim.x + threadIdx.x) * 4;
    int row = (int)(i4 >> 14);                   // NPIX = 16384 per row
    float m = mx[row];
    float4* p = (float4*)(kws + i4);
    float4 v = *p;
    v.x = __expf(v.x - m); v.y = __expf(v.y - m);
    v.z = __expf(v.z - m); v.w = __expf(v.w - m);
    *p = v;
}

// ---------------------------------------------------------------- kernel 3
// yk[b,c,r] += sum_{n in chunk} y[b,c,n] * ek[b,r,n]   (ek precomputed in kws)
// grid (8 kchunks, BATCH), block 512 (16 waves). wave = m-tile (16 channels).
__global__ void yk_accum(const float* __restrict__ y,
                         const float* __restrict__ ek,
                         float* __restrict__ yk) {
    const int lane = threadIdx.x & 31;
    const int wave = threadIdx.x >> 5;
    const int l16  = lane & 15;
    const int half = lane >> 4;
    const int b    = blockIdx.y;
    const int c0   = wave * 16;
    const int kbase = blockIdx.x * (NPIX / 8);   // 2048 per chunk

    const float* yb = y + (size_t)b * DIMC * NPIX;
    const float* kr = ek + (size_t)b * RED * NPIX;

    v8f acc0 = {}, acc1 = {};

    for (int k = kbase; k < kbase + NPIX / 8; k += 4) {
        v2f a;
        const float* ap = yb + (size_t)(c0 + l16) * NPIX + k + 2 * half;
        a.x = ap[0]; a.y = ap[1];

        v2f b0, b1;
        const float* kpA = kr + (size_t)l16 * NPIX + k + 2 * half;
        const float* kpB = kr + (size_t)(16 + l16) * NPIX + k + 2 * half;
        b0.x = kpA[0]; b0.y = kpA[1];
        b1.x = kpB[0]; b1.y = kpB[1];

        acc0 = wmma4(a, b0, acc0);
        acc1 = wmma4(a, b1, acc1);
    }

    #pragma unroll
    for (int vi = 0; vi < 8; ++vi) {
        int c = c0 + vi + 8 * half;
        atomicAdd(&yk[((size_t)b * DIMC + c) * RED + l16],      acc0[vi]);
        atomicAdd(&yk[((size_t)b * DIMC + c) * RED + 16 + l16], acc1[vi]);
    }
}

// ---------------------------------------------------------------- kernel 4a
// ctx[b,c,r] = (v_w @ yk_b)[c,r]/sumek[r] + v_b[c];  bias[b,c] = g*sum_r ctx*q_b[r]
// grid BATCH, block 512 (16 waves = 16 m-tiles), K=256.
__global__ void ctx_kernel(const float* __restrict__ vw,
                           const float* __restrict__ vb,
                           const float* __restrict__ qb,
                           const float* __restrict__ gamma,
                           const float* __restrict__ yk,
                           const float* __restrict__ sm,
                           float* __restrict__ ctx,
                           float* __restrict__ bias) {
    const int lane = threadIdx.x & 31;
    const int wave = threadIdx.x >> 5;
    const int l16  = lane & 15;
    const int half = lane >> 4;
    const int b    = blockIdx.x;
    const int c0   = wave * 16;
    const size_t bD = (size_t)b * DIMC;

    v8f acc0 = {}, acc1 = {};
    for (int k = 0; k < DIMC; k += 4) {
        v2f a;
        const float* ap = vw + (size_t)(c0 + l16) * DIMC + k + 2 * half;
        a.x = ap[0]; a.y = ap[1];

        v2f b0, b1;
        const float* bp0 = yk + (bD + k + 2 * half) * RED + l16;
        const float* bp1 = yk + (bD + k + 2 * half) * RED + 16 + l16;
        b0.x = bp0[0]; b0.y = bp0[RED];
        b1.x = bp1[0]; b1.y = bp1[RED];

        acc0 = wmma4(a, b0, acc0);
        acc1 = wmma4(a, b1, acc1);
    }

    const float g    = gamma[0];
    const float smA  = sm[b * RED + l16];
    const float smB  = sm[b * RED + 16 + l16];
    const float qbA  = qb[l16];
    const float qbB  = qb[16 + l16];

    #pragma unroll
    for (int vi = 0; vi < 8; ++vi) {
        int c = c0 + vi + 8 * half;
        float cv0 = acc0[vi] / smA + vb[c];
        float cv1 = acc1[vi] / smB + vb[c];
        ctx[(bD + c) * RED + l16]      = cv0;
        ctx[(bD + c) * RED + 16 + l16] = cv1;

        float p = cv0 * qbA + cv1 * qbB;
        #pragma unroll
        for (int off = 1; off < 16; off <<= 1)
            p += __shfl_xor(p, off, 32);     // reduce within each 16-lane half
        if (l16 == 0) bias[bD + c] = g * p;
    }
}

// ---------------------------------------------------------------- kernel 4b
// M[b,c,c2] = g * sum_r ctx[b,c,r]*q_w[r,c2]
// grid BATCH, block 512 (16 waves = m-tiles), loop 16 n-tiles, K=32.
__global__ void mout_kernel(const float* __restrict__ ctx,
                            const float* __restrict__ qw,
                            const float* __restrict__ gamma,
                            float* __restrict__ Mout) {
    const int lane = threadIdx.x & 31;
    const int wave = threadIdx.x >> 5;
    const int l16  = lane & 15;
    const int half = lane >> 4;
    const int b    = blockIdx.x;
    const int m0   = wave * 16;
    const size_t bD = (size_t)b * DIMC;
    const float g  = gamma[0];

    for (int nt = 0; nt < 16; ++nt) {
        v8f acc = {};
        for (int k = 0; k < RED; k += 4) {
            v2f a;
            const float* ap = ctx + (bD + m0 + l16) * RED + k + 2 * half;
            a.x = ap[0]; a.y = ap[1];
            v2f bf;
            const float* bp = qw + (size_t)(k + 2 * half) * DIMC + nt * 16 + l16;
            bf.x = bp[0]; bf.y = bp[DIMC];
            acc = wmma4(a, bf, acc);
        }
        #pragma unroll
        for (int vi = 0; vi < 8; ++vi) {
            int c = m0 + vi + 8 * half;
            Mout[(bD + c) * DIMC + nt * 16 + l16] = g * acc[vi];
        }
    }
}

// ---------------------------------------------------------------- kernel 5
// out[b,c,n] = M_b[c,:] @ x[b,:,n] + bias[b,c] + x[b,c,n]
// grid (NPIX/64, BATCH), block 512 (16 waves = 16 m-tiles covering all 256 c).
// The 256x64 x-slab is staged once into LDS with async b128 copies and then
// serves all 16 waves' B fragments plus the residual term.
__global__ void out_kernel(const float* __restrict__ x,
                           const float* __restrict__ Mout,
                           const float* __restrict__ bias,
                           float* __restrict__ out) {
    __shared__ float xs[DIMC][64];

    const int lane = threadIdx.x & 31;
    const int wave = threadIdx.x >> 5;
    const int l16  = lane & 15;
    const int half = lane >> 4;
    const int b    = blockIdx.y;
    const int c0   = wave * 16;
    const int n0   = blockIdx.x * 64;
    const size_t bD = (size_t)b * DIMC;

    const float* xb = x + bD * NPIX;

    // ---- stage x[:, n0:n0+64] into LDS: 512 threads x 8 rounds x 16B
    {
        const int row0 = threadIdx.x >> 4;        // 0..31
        const int cg   = (threadIdx.x & 15) * 4;  // column group (4 floats)
        #pragma unroll
        for (int i = 0; i < 8; ++i) {
            const int row = row0 + i * 32;
            const float* g = xb + (size_t)row * NPIX + n0 + cg;
            unsigned lds_off = (unsigned)(size_t)&xs[row][cg];
            async_load_b128(lds_off, g);
        }
    }
    wait_async0();
    __syncthreads();

    v8f acc[4];
    #pragma unroll
    for (int nt = 0; nt < 4; ++nt) acc[nt] = {};

    for (int k = 0; k < DIMC; k += 4) {
        v2f a;
        const float* ap = Mout + (bD + c0 + l16) * DIMC + k + 2 * half;
        a.x = ap[0]; a.y = ap[1];

        v2f bf[4];
        #pragma unroll
        for (int nt = 0; nt < 4; ++nt) {
            bf[nt].x = xs[k + 2 * half][nt * 16 + l16];
            bf[nt].y = xs[k + 2 * half + 1][nt * 16 + l16];
        }
        #pragma unroll
        for (int nt = 0; nt < 4; ++nt)
            acc[nt] = wmma4(a, bf[nt], acc[nt]);
    }

    #pragma unroll
    for (int nt = 0; nt < 4; ++nt)
        #pragma unroll
        for (int vi = 0; vi < 8; ++vi) {
            int c   = c0 + vi + 8 * half;
            int col = nt * 16 + l16;
            out[(bD + c) * NPIX + n0 + col] = acc[nt][vi] + bias[bD + c] + xs[c][col];
        }
}

// ================================================================ launch
extern "C" void kernel_launch(void* const* d_in, const int* in_sizes, int n_in,
                              void* d_out, int out_size, void* d_ws, size_t ws_size,
                              hipStream_t stream) {
    const float* x     = (const float*)d_in[0];
    const float* y     = (const float*)d_in[1];
    const float* q_w   = (const float*)d_in[2];
    const float* q_b   = (const float*)d_in[3];
    const float* k_w   = (const float*)d_in[4];
    const float* k_b   = (const float*)d_in[5];
    const float* v_w   = (const float*)d_in[6];
    const float* v_b   = (const float*)d_in[7];
    const float* gamma = (const float*)d_in[8];
    float* out = (float*)d_out;

    float* ws   = (float*)d_ws;
    float* kws  = ws;                                 // B*RED*NPIX = 8388608
    float* mx   = kws  + (size_t)BATCH * RED * NPIX;  // 512
    float* sm   = mx   + BATCH * RED;                 // 512
    float* yk   = sm   + BATCH * RED;                 // B*DIMC*RED = 131072
    float* ctx  = yk   + (size_t)BATCH * DIMC * RED;  // 131072
    float* Mo   = ctx  + (size_t)BATCH * DIMC * RED;  // B*DIMC*DIMC = 1048576
    float* bias = Mo   + (size_t)BATCH * DIMC * DIMC; // 4096

    // 0) zero the atomic accumulation buffer
    zero_f32<<<(BATCH * DIMC * RED + 255) / 256, 256, 0, stream>>>(yk, BATCH * DIMC * RED);

    // 1) k logits = k_w @ y + k_b
    k_project<<<dim3(NPIX / 512, BATCH), 256, 0, stream>>>(y, k_w, k_b, kws);

    // 2) per-row softmax stats
    softmax_stats<<<BATCH * RED, 256, 0, stream>>>(kws, mx, sm);

    // 2.5) k <- exp(k - max) in place (removes all redundant exp from kernel 3)
    exp_inplace<<<(BATCH * RED * NPIX / 4) / 256, 256, 0, stream>>>(kws, mx);

    // 3) yk = Y @ ek^T  (K-split, atomic accumulate)
    yk_accum<<<dim3(8, BATCH), 512, 0, stream>>>(y, kws, yk);

    // 4a) ctx = v_w @ yk / sumek + v_b ; bias = gamma * ctx @ q_b
    ctx_kernel<<<BATCH, 512, 0, stream>>>(v_w, v_b, q_b, gamma, yk, sm, ctx, bias);

    // 4b) M = gamma * ctx @ q_w
    mout_kernel<<<BATCH, 512, 0, stream>>>(ctx, q_w, gamma, Mo);

    // 5) out = M @ x + bias + x  (x slab staged via async-to-LDS)
    out_kernel<<<dim3(NPIX / 64, BATCH), 512, 0, stream>>>(x, Mo, bias, out);
}